// Net_41772851920951
// MI455X (gfx1250) — compile-verified
//
#include <hip/hip_runtime.h>
#include <hip/hip_bf16.h>

typedef __attribute__((ext_vector_type(2))) float v2f;
typedef __attribute__((ext_vector_type(8))) float v8f;

#define WG_GEMM 128

// ---------------------------------------------------------------------------
// WMMA fp32 GEMM: Y[n x F] = X[n x K] * W[K x F] (+ bias), K multiple of 4,
// K <= 96. NT = ceil(F/16) as template constant. One block = 4 waves = 64 rows.
// Uses V_WMMA_F32_16X16X4_F32 (exact fp32, matches fp32 reference).
// ---------------------------------------------------------------------------
template <int NT>
__global__ __launch_bounds__(WG_GEMM) void gemm_wmma_f32(
    const float* __restrict__ X, const float* __restrict__ W,
    const float* __restrict__ bias, float* __restrict__ Y, int n, int K,
    int F) {
  constexpr int NTC = NT * 16;
  __shared__ float ldsW[96 * NTC];

  const int tid = threadIdx.x;
  // Stage W into LDS, zero-padded to NTC columns.
  for (int idx = tid; idx < K * NTC; idx += WG_GEMM) {
    const int k = idx / NTC;
    const int c = idx - k * NTC;
    ldsW[idx] = (c < F) ? W[k * F + c] : 0.0f;
  }
  __syncthreads();

  const int wave = tid >> 5;
  const int lane = tid & 31;
  const int half = lane >> 4;  // 0: lanes 0-15, 1: lanes 16-31
  const int l16 = lane & 15;
  const int m_base = blockIdx.x * 64 + wave * 16;

  // A-fragment row for this lane (clamped so all lanes stay active: EXEC must
  // be all-ones across the WMMA).
  int arow = m_base + l16;
  if (arow >= n) arow = n - 1;
  const float* __restrict__ xr = X + (size_t)arow * K;

  v8f acc[NT] = {};

  for (int kk = 0; kk < K; kk += 4) {
    // A 16x4 layout: lane L holds K = kk + 2*(L/16) + {0,1}
    const int k0 = kk + 2 * half;
    v2f a;
    a.x = xr[k0];
    a.y = xr[k0 + 1];
#pragma unroll
    for (int t = 0; t < NT; ++t) {
      // B 4x16 layout: lane L holds rows K = kk + 2*(L/16) + {0,1}, col L%16
      const int col = t * 16 + l16;
      v2f b;
      b.x = ldsW[k0 * NTC + col];
      b.y = ldsW[(k0 + 1) * NTC + col];
      acc[t] = __builtin_amdgcn_wmma_f32_16x16x4_f32(
          false, a, false, b, (short)0, acc[t], false, false);
    }
  }

  // C/D layout: VGPR r, lane L -> row m_base + r + 8*(L/16), col L%16
#pragma unroll
  for (int t = 0; t < NT; ++t) {
    const int col = t * 16 + l16;
    if (col < F) {
      const float bv = bias ? bias[col] : 0.0f;
#pragma unroll
      for (int r = 0; r < 8; ++r) {
        const int m = m_base + r + 8 * half;
        if (m < n) Y[(size_t)m * F + col] = acc[t][r] + bv;
      }
    }
  }
}

// ---------------------------------------------------------------------------
// Degree / normalization
// ---------------------------------------------------------------------------
__global__ void k_fill1(float* __restrict__ deg, int n) {
  const int i = blockIdx.x * blockDim.x + threadIdx.x;
  if (i < n) deg[i] = 1.0f;  // self-loop contribution
}

__global__ void k_deg(const int* __restrict__ dst, float* __restrict__ deg,
                      int E) {
  const int e = blockIdx.x * blockDim.x + threadIdx.x;
  if (e < E) atomicAdd(&deg[dst[e]], 1.0f);
}

__global__ void k_dinv(const float* __restrict__ deg, float* __restrict__ dinv,
                       int n) {
  const int i = blockIdx.x * blockDim.x + threadIdx.x;
  if (i < n) dinv[i] = rsqrtf(deg[i]);  // deg >= 1 always (self-loops)
}

// ---------------------------------------------------------------------------
// Aggregation: out[i] = b + h[i]*dinv[i]^2 (self-loop)  then scatter edges.
// ---------------------------------------------------------------------------
__global__ void k_agg_init(const float* __restrict__ h,
                           const float* __restrict__ dinv,
                           const float* __restrict__ b, float* __restrict__ out,
                           int n, int F) {
  const long long idx = (long long)blockIdx.x * blockDim.x + threadIdx.x;
  if (idx < (long long)n * F) {
    const int i = (int)(idx / F);
    const int f = (int)(idx - (long long)i * F);
    const float di = dinv[i];
    out[idx] = b[f] + h[idx] * di * di;
  }
}

// One wave32 per edge: 96 features -> 3 strided atomics per lane.
__global__ void k_agg_edges(const int* __restrict__ src,
                            const int* __restrict__ dst,
                            const float* __restrict__ h,
                            const float* __restrict__ dinv,
                            float* __restrict__ out, int E, int F) {
  const int e = blockIdx.x * 8 + (threadIdx.x >> 5);
  const int lane = threadIdx.x & 31;
  if (e >= E) return;
  const int s = src[e];
  const int d = dst[e];
  const float c = dinv[s] * dinv[d];
  const float* __restrict__ hr = h + (size_t)s * F;
  float* __restrict__ orow = out + (size_t)d * F;
  for (int f = lane; f < F; f += 32) atomicAdd(&orow[f], hr[f] * c);
}

__global__ void k_relu(float* __restrict__ x, long long n) {
  const long long i = (long long)blockIdx.x * blockDim.x + threadIdx.x;
  if (i < n) x[i] = fmaxf(x[i], 0.0f);
}

// ---------------------------------------------------------------------------
// Launch: deg -> dinv -> [GEMM -> agg_init -> agg_edges -> relu] x2 -> head
// ---------------------------------------------------------------------------
extern "C" void kernel_launch(void* const* d_in, const int* in_sizes, int n_in,
                              void* d_out, int out_size, void* d_ws,
                              size_t ws_size, hipStream_t stream) {
  const float* x = (const float*)d_in[0];
  const int* ei = (const int*)d_in[1];
  const float* W1 = (const float*)d_in[2];
  const float* b1 = (const float*)d_in[3];
  const float* W2 = (const float*)d_in[4];
  const float* b2 = (const float*)d_in[5];
  const float* Wl = (const float*)d_in[6];
  const float* bl = (const float*)d_in[7];
  float* out = (float*)d_out;

  const int dh = in_sizes[3];        // 96
  const int din = in_sizes[2] / dh;  // 96
  const int ncls = in_sizes[7];      // 40
  const int n = in_sizes[0] / din;   // 50000
  const int E = in_sizes[1] / 2;     // 800000
  const int* src = ei;
  const int* dst = ei + E;

  float* ws = (float*)d_ws;
  float* deg = ws;
  float* dinv = ws + n;
  float* H = ws + 2 * (size_t)n;       // [n x dh] pre-aggregation features
  float* A = H + (size_t)n * dh;       // [n x dh] aggregated features

  const int t256n = (n + 255) / 256;
  const int t256e = (E + 255) / 256;
  const long long nf = (long long)n * dh;
  const int t256nf = (int)((nf + 255) / 256);
  const int gblk = (n + 63) / 64;
  const int eblk = (E + 7) / 8;

  // Normalization coefficients (shared by both conv layers).
  k_fill1<<<t256n, 256, 0, stream>>>(deg, n);
  k_deg<<<t256e, 256, 0, stream>>>(dst, deg, E);
  k_dinv<<<t256n, 256, 0, stream>>>(deg, dinv, n);

  // Layer 1: H = X @ W1 ; A = D^-1/2 (A+I) D^-1/2 H + b1 ; relu
  gemm_wmma_f32<6><<<gblk, WG_GEMM, 0, stream>>>(x, W1, nullptr, H, n, din, dh);
  k_agg_init<<<t256nf, 256, 0, stream>>>(H, dinv, b1, A, n, dh);
  k_agg_edges<<<eblk, 256, 0, stream>>>(src, dst, H, dinv, A, E, dh);
  k_relu<<<t256nf, 256, 0, stream>>>(A, nf);

  // Layer 2: H = A @ W2 ; A = aggregate ; relu   (buffers ping-pong safely)
  gemm_wmma_f32<6><<<gblk, WG_GEMM, 0, stream>>>(A, W2, nullptr, H, n, dh, dh);
  k_agg_init<<<t256nf, 256, 0, stream>>>(H, dinv, b2, A, n, dh);
  k_agg_edges<<<eblk, 256, 0, stream>>>(src, dst, H, dinv, A, E, dh);
  k_relu<<<t256nf, 256, 0, stream>>>(A, nf);

  // Classifier head: out = A @ Wl + bl  (F=40 -> NT=3 with column guards)
  gemm_wmma_f32<3><<<gblk, WG_GEMM, 0, stream>>>(A, Wl, bl, out, n, dh, ncls);
}